// Histogram_63668595196221
// MI455X (gfx1250) — compile-verified
//
#include <hip/hip_runtime.h>
#include <hip/hip_bf16.h>
#include <stdint.h>

namespace {
constexpr int kBatch    = 64;
constexpr int kNPoints  = 8192;
constexpr int kFeat     = 64;
constexpr int kBins     = 10;
constexpr int kBinsPad  = 17;   // odd word stride -> spreads LDS banks
constexpr int kNSub     = 64;   // replicated sub-histograms (tid % 64)
constexpr int kWGS      = 256;  // 8 wave32 per workgroup
constexpr int kWgPerB   = 16;
constexpr int kElemsPerB  = kNPoints * kFeat;           // 524288
constexpr int kElemsPerWG = kElemsPerB / kWgPerB;       // 32768 floats = 128KB
constexpr int kTileElems  = kWGS * 4;                   // 1024 floats = 4KB tile
constexpr int kTileBytes  = kTileElems * 4;             // 4096
constexpr int kNTiles     = kElemsPerWG / kTileElems;   // 32
constexpr int kNBuf       = 4;                          // LDS ring buffers
constexpr int kDepth      = 3;                          // async ops in flight
}

// Streams 128KB/WG through a 4-deep ring of LDS tiles filled by the CDNA5
// async global->LDS engine (3 ops in flight per wave, ASYNCcnt-tracked),
// accumulates into 64 bank-spread replicated LDS histograms via ds_add_u32,
// and emits exact integer partial counts per WG.
__global__ __launch_bounds__(kWGS)
void hist_main(const float* __restrict__ in, unsigned* __restrict__ partial) {
  __shared__ __attribute__((aligned(16))) float tile[kNBuf][kTileElems]; // 16KB
  __shared__ unsigned hist[kNSub * kBinsPad];                            // 4.25KB

  const int tid = threadIdx.x;
  const int wg  = blockIdx.x;                     // 0 .. kBatch*kWgPerB-1
  unsigned* const myhist = &hist[(tid & (kNSub - 1)) * kBinsPad];

  const float* base = in + (size_t)wg * kElemsPerWG;

  // Low 32 bits of a generic shared-space pointer are the LDS byte offset
  const unsigned lbase = (unsigned)(uint64_t)(uintptr_t)(&tile[0][tid * 4]);
  const unsigned vbase = (unsigned)(tid * 16);    // global byte offset of lane slot

  // Prime the pipeline first so global fetch latency hides behind LDS zeroing.
#pragma unroll
  for (int t = 0; t < kDepth; ++t) {
    asm volatile("global_load_async_to_lds_b128 %0, %1, %2"
                 :: "v"(lbase + t * kTileBytes),
                    "v"(vbase + (unsigned)(t * kTileBytes)),
                    "s"(base) : "memory");
  }

  for (int i = tid; i < kNSub * kBinsPad; i += kWGS) hist[i] = 0u;
  __syncthreads();   // drains DScnt; does not drain ASYNCcnt

#pragma unroll
  for (int t = 0; t < kNTiles; ++t) {
    if (t + kDepth < kNTiles) {
      // Refill buf[(t+kDepth)&3] == buf[(t-1)&3]. Its readers ran last
      // iteration; drain DScnt so the async LDS write cannot pass them
      // (async engine writes are not ordered against this wave's ds ops).
      const int u = t + kDepth;
      asm volatile("s_wait_dscnt 0x0" ::: "memory");
      asm volatile("global_load_async_to_lds_b128 %0, %1, %2"
                   :: "v"(lbase + (u & (kNBuf - 1)) * kTileBytes),
                      "v"(vbase + (unsigned)(u * kTileBytes)),
                      "s"(base) : "memory");
      // 4 in flight after issue; oldest (tile t) is complete at <=3.
      asm volatile("s_wait_asynccnt 0x3" ::: "memory");
    } else if (t == kNTiles - 3) {
      asm volatile("s_wait_asynccnt 0x2" ::: "memory");
    } else if (t == kNTiles - 2) {
      asm volatile("s_wait_asynccnt 0x1" ::: "memory");
    } else {
      asm volatile("s_wait_asynccnt 0x0" ::: "memory");
    }

    const float4 v = *(const float4*)(&tile[t & (kNBuf - 1)][tid * 4]); // ds_load_b128
    const float xs[4] = {v.x, v.y, v.z, v.w};
#pragma unroll
    for (int j = 0; j < 4; ++j) {
      int b = (int)(xs[j] * 10.0f);            // same f32 math as the reference
      b = b < (kBins - 1) ? b : (kBins - 1);   // defensive clamp (x in [0,1))
      atomicAdd(&myhist[b], 1u);               // ds_add_u32, no return
    }
  }

  __syncthreads();
  // Threads 0..9 fold the 64 replicas into this WG's partial counts.
  for (int b = tid; b < kBins; b += kWGS) {
    unsigned s = 0;
#pragma unroll 8
    for (int r = 0; r < kNSub; ++r) s += hist[r * kBinsPad + b];
    partial[wg * kBins + b] = s;   // every slot written -> deterministic
  }
}

// 640 outputs: out[b][k] = weight[k] * sum of the 16 WG partials of batch b.
__global__ __launch_bounds__(320)
void hist_finalize(const unsigned* __restrict__ partial,
                   const float* __restrict__ weight,
                   float* __restrict__ out) {
  const int i = blockIdx.x * blockDim.x + threadIdx.x;
  if (i >= kBatch * kBins) return;
  const int b = i / kBins;
  const int k = i - b * kBins;
  unsigned s = 0;
#pragma unroll
  for (int w = 0; w < kWgPerB; ++w)
    s += partial[(b * kWgPerB + w) * kBins + k];
  out[i] = (float)s * weight[k];   // counts <= 524288: exact in f32
}

extern "C" void kernel_launch(void* const* d_in, const int* in_sizes, int n_in,
                              void* d_out, int out_size, void* d_ws, size_t ws_size,
                              hipStream_t stream) {
  const float* x = (const float*)d_in[0];   // (64, 8192, 64) f32
  const float* w = (const float*)d_in[1];   // (10,) f32
  float* out = (float*)d_out;               // (64, 10) f32
  unsigned* partial = (unsigned*)d_ws;      // 1024 * 10 u32 = 40KB scratch

  hipLaunchKernelGGL(hist_main, dim3(kBatch * kWgPerB), dim3(kWGS), 0, stream,
                     x, partial);
  hipLaunchKernelGGL(hist_finalize, dim3(2), dim3(320), 0, stream,
                     partial, w, out);
}